// TokenSelfAttend_16381005267527
// MI455X (gfx1250) — compile-verified
//
#include <hip/hip_runtime.h>
#include <cmath>

#define BB 8
#define TT 128
#define DD 768
#define HH 768
#define LN_EPS 1e-5f
#define LDSW (768 + 4)   // padded LDS row stride (floats) -> conflict-free ds_load_b64

typedef __attribute__((ext_vector_type(2))) float v2f;
typedef __attribute__((ext_vector_type(8))) float v8f;

__device__ __forceinline__ float gelu_erf(float v) {
    return 0.5f * v * (1.0f + erff(v * 0.70710678118654752f));
}

// ---------------------------------------------------------------------------
// Kernel 1: LayerNorm + per-token GEMM1 (xn @ W1[t] + b1[t]) + exact GELU
// grid = (H/128, T), block = 256 (8 waves). Wave w owns one 16-col N tile.
// A-matrix = xn (8 rows padded to 16; pad rows are never read back), streamed
// from LDS. B-matrix = W1[t] streamed once from HBM straight into fragments.
// ---------------------------------------------------------------------------
__global__ __launch_bounds__(256) void ffn_ln_gemm1_kernel(
    const float* __restrict__ x, const float* __restrict__ gamma,
    const float* __restrict__ beta, const float* __restrict__ W1,
    const float* __restrict__ b1, float* __restrict__ hws) {
  __shared__ float xn[16 * LDSW];

  const int lane = threadIdx.x & 31;
  const int wave = threadIdx.x >> 5;          // 0..7 == batch row in LN phase
  const int t    = blockIdx.y;
  const int n0   = blockIdx.x * 128 + wave * 16;

  // ---- LayerNorm: wave w normalizes batch row w of this token ----
  {
    const float* xrow = x + ((size_t)wave * TT + t) * DD;
    float s = 0.f, s2 = 0.f;
    for (int j = lane; j < DD; j += 32) {
      float v = xrow[j];
      s += v; s2 += v * v;
    }
    for (int m = 16; m > 0; m >>= 1) {          // wave32 tree reduction
      s  += __shfl_xor(s, m, 32);
      s2 += __shfl_xor(s2, m, 32);
    }
    const float mu   = s * (1.0f / DD);
    const float var  = s2 * (1.0f / DD) - mu * mu;
    const float rstd = rsqrtf(var + LN_EPS);
    float* dst = xn + wave * LDSW;
    const float* g  = gamma + (size_t)t * DD;
    const float* bt = beta  + (size_t)t * DD;
    for (int j = lane; j < DD; j += 32)
      dst[j] = (xrow[j] - mu) * rstd * g[j] + bt[j];
  }
  __syncthreads();

  // ---- GEMM1: 16x16 tile per wave, K=768 in steps of 4 (WMMA f32 16x16x4) ----
  const int half  = lane >> 4;     // 0: K+0/K+1 lanes, 1: K+2/K+3 lanes
  const int ksel  = half << 1;
  const int nlane = lane & 15;

  const float* bp = W1 + (size_t)t * DD * HH + (size_t)ksel * HH + (n0 + nlane);
  const float* ap = xn + nlane * LDSW + ksel;

  v8f acc = {};
#pragma unroll 4
  for (int k = 0; k < DD; k += 4) {
    v2f a; a.x = ap[0]; a.y = ap[1];           // ds_load_b64 from LDS
    v2f b; b.x = bp[0]; b.y = bp[HH];          // stream W1 once from HBM
    ap += 4;
    bp += (size_t)4 * HH;
    acc = __builtin_amdgcn_wmma_f32_16x16x4_f32(
        false, a, false, b, (short)0, acc, false, false);
  }

  // ---- bias + GELU, store rows 0..7 (lanes 0-15); pad rows discarded ----
  if (lane < 16) {
    const float bias = b1[(size_t)t * HH + n0 + nlane];
    float* hdst = hws + (size_t)t * HH + (n0 + nlane);
#pragma unroll
    for (int r = 0; r < 8; ++r)
      hdst[(size_t)r * TT * HH] = gelu_erf(acc[r] + bias);
  }
}

// ---------------------------------------------------------------------------
// Kernel 2: per-token GEMM2 (h @ W2[t] + b2[t]) + residual
// grid = (D/128, T), block = 256. Same tiling; h reloaded from workspace into
// LDS (pad rows zeroed), W2 streamed once from HBM.
// ---------------------------------------------------------------------------
__global__ __launch_bounds__(256) void ffn_gemm2_kernel(
    const float* __restrict__ hws, const float* __restrict__ W2,
    const float* __restrict__ b2, const float* __restrict__ x,
    float* __restrict__ out) {
  __shared__ float hl[16 * LDSW];

  const int lane = threadIdx.x & 31;
  const int wave = threadIdx.x >> 5;
  const int t    = blockIdx.y;
  const int n0   = blockIdx.x * 128 + wave * 16;

  // ---- stage h[b=wave, t, :] into LDS, zero the pad rows ----
  {
    const float* hsrc = hws + ((size_t)wave * TT + t) * HH;
    float* d0 = hl + wave * LDSW;
    float* d1 = hl + (wave + 8) * LDSW;
    for (int j = lane; j < HH; j += 32) {
      d0[j] = hsrc[j];
      d1[j] = 0.0f;
    }
  }
  __syncthreads();

  const int half  = lane >> 4;
  const int ksel  = half << 1;
  const int nlane = lane & 15;

  const float* bp = W2 + (size_t)t * HH * DD + (size_t)ksel * DD + (n0 + nlane);
  const float* ap = hl + nlane * LDSW + ksel;

  v8f acc = {};
#pragma unroll 4
  for (int k = 0; k < HH; k += 4) {
    v2f a; a.x = ap[0]; a.y = ap[1];
    v2f b; b.x = bp[0]; b.y = bp[DD];
    ap += 4;
    bp += (size_t)4 * DD;
    acc = __builtin_amdgcn_wmma_f32_16x16x4_f32(
        false, a, false, b, (short)0, acc, false, false);
  }

  // ---- bias + residual + store rows 0..7 ----
  if (lane < 16) {
    const int col = n0 + nlane;
    const float bias = b2[(size_t)t * DD + col];
#pragma unroll
    for (int r = 0; r < 8; ++r) {
      const size_t idx = (size_t)r * TT * DD + (size_t)t * DD + col;
      out[idx] = acc[r] + bias + x[idx];
    }
  }
}

extern "C" void kernel_launch(void* const* d_in, const int* in_sizes, int n_in,
                              void* d_out, int out_size, void* d_ws, size_t ws_size,
                              hipStream_t stream) {
  const float* x     = (const float*)d_in[0];
  const float* gamma = (const float*)d_in[1];
  const float* beta  = (const float*)d_in[2];
  const float* W1    = (const float*)d_in[3];
  const float* b1    = (const float*)d_in[4];
  const float* W2    = (const float*)d_in[5];
  const float* b2    = (const float*)d_in[6];
  float* out = (float*)d_out;
  float* hws = (float*)d_ws;  // B*T*H floats = 3 MB intermediate h

  dim3 block(256);
  dim3 grid1(HH / 128, TT);   // (6, 128) = 768 workgroups
  dim3 grid2(DD / 128, TT);

  ffn_ln_gemm1_kernel<<<grid1, block, 0, stream>>>(x, gamma, beta, W1, b1, hws);
  ffn_gemm2_kernel<<<grid2, block, 0, stream>>>(hws, W2, b2, x, out);
}